// Bind_41532333752518
// MI455X (gfx1250) — compile-verified
//
#include <hip/hip_runtime.h>
#include <hip/hip_bf16.h>

typedef __attribute__((ext_vector_type(2))) float v2f;
typedef __attribute__((ext_vector_type(8))) float v8f;

#define D_DIM 128
#define H_HEADS 8

// ---------------------------------------------------------------------------
// helpers
// ---------------------------------------------------------------------------
__device__ __forceinline__ float wave_sum32(float v) {
    v += __shfl_xor(v, 16, 32);
    v += __shfl_xor(v, 8, 32);
    v += __shfl_xor(v, 4, 32);
    v += __shfl_xor(v, 2, 32);
    v += __shfl_xor(v, 1, 32);
    return v;
}

// monotone float <-> uint mapping so we can use integer atomicMax for f32 max
__device__ __forceinline__ unsigned fflip(float f) {
    unsigned u = __float_as_uint(f);
    return (u & 0x80000000u) ? ~u : (u | 0x80000000u);
}
__device__ __forceinline__ float funflip(unsigned u) {
    unsigned b = (u & 0x80000000u) ? (u & 0x7FFFFFFFu) : ~u;
    return __uint_as_float(b);
}

// ---------------------------------------------------------------------------
// Tiled fp32 GEMM on V_WMMA_F32_16X16X4_F32.
// C[M,Nfull] = A[M,K] @ B[K,Nfull]  (optionally ReLU'ed).
// Block: 256 threads = 8 waves; block tile = 128 rows x 128 cols.
// Wave w owns rows [row0+16w, +16); its 8 accumulators cover the 128 cols.
//
// B is staged through LDS in 128-deep K chunks (64 KB) using CDNA5
// GLOBAL_LOAD_ASYNC_TO_LDS_B32 (ASYNCcnt-tracked, no VGPR round-trip).
// The async copy applies a k-pair swizzle on the fly:
//     element (k, n)  ->  sB[(k>>1)*256 + n*2 + (k&1)]
// so each WMMA B-fragment {B[k][n], B[k+1][n]} is one aligned ds_load_b64
// into an adjacent VGPR pair (no v_mov re-pairing in the inner loop).
//
// NOTE: the LDS destination operand is formed by truncating the generic
// address of the sB element to 32 bits (flat LDS aperture keeps the LDS byte
// offset in addr[31:0]). This also *captures* sB into the asm so the
// "memory" clobbers cover it — without that, alias analysis deletes the
// B-fragment loads entirely (observed in the previous round's disasm).
// ---------------------------------------------------------------------------
__global__ void gemm_wmma_f32(const float* __restrict__ A,
                              const float* __restrict__ B,
                              float* __restrict__ C,
                              int M, int K, int Nfull, int relu) {
    __shared__ float sB[128 * 128];  // 64 KB

    const int tid  = threadIdx.x;
    const int wave = tid >> 5;
    const int lane = tid & 31;
    const int row0 = blockIdx.x * 128 + wave * 16;
    const int col0 = blockIdx.y * 128;

    const int a_row   = row0 + (lane & 15);
    const int a_row_c = (a_row < M) ? a_row : (M - 1);   // clamp (EXEC stays full)
    const int koff    = (lane >> 4) * 2;                 // 0 for lanes 0-15, 2 for 16-31
    const int bn      = lane & 15;

    const v2f* sBp = (const v2f*)sB;                     // pair view: [64][128] x v2f

    v8f acc[8] = {};

    for (int kc = 0; kc < K; kc += 128) {
        // ---- async global->LDS staging with k-pair swizzle (uniform trip count)
        for (int idx = tid; idx < 128 * 128; idx += 256) {
            const int k = idx >> 7;   // row within chunk
            const int n = idx & 127;  // col within tile
            // swizzled element index inside sB
            const int el = (((k >> 1) * 128 + n) << 1) + (k & 1);
            // LDS byte offset == low 32 bits of the generic address of &sB[el];
            // taking the address also captures sB for the asm memory clobber.
            const unsigned ldsoff = (unsigned)(size_t)(const void*)&sB[el];
            const float* gp = &B[(size_t)(kc + k) * Nfull + col0 + n];
            asm volatile("global_load_async_to_lds_b32 %0, %1, off"
                         :: "v"(ldsoff), "v"(gp)
                         : "memory");
        }
        asm volatile("s_wait_asynccnt 0x0" ::: "memory");
        __syncthreads();

        // ---- WMMA inner loop
        for (int k = 0; k < 128; k += 4) {
            v2f a;
            const size_t abase = (size_t)a_row_c * K + kc + k + koff;
            a.x = A[abase];
            a.y = A[abase + 1];
            const int prow = ((k + koff) >> 1) * 128 + bn;
#pragma unroll
            for (int t = 0; t < 8; ++t) {
                const v2f b = sBp[prow + t * 16];
                acc[t] = __builtin_amdgcn_wmma_f32_16x16x4_f32(
                    false, a, false, b, (short)0, acc[t], false, false);
            }
        }
        __syncthreads();
    }

    // store: VGPR j -> M = row0 + j + 8*(lane>=16), N = col0 + 16t + (lane&15)
    const int rbase = row0 + (lane >> 4) * 8;
#pragma unroll
    for (int t = 0; t < 8; ++t) {
        const int col = col0 + t * 16 + bn;
#pragma unroll
        for (int j = 0; j < 8; ++j) {
            const int r = rbase + j;
            if (r < M) {
                float val = acc[t][j];
                if (relu) val = fmaxf(val, 0.0f);
                C[(size_t)r * Nfull + col] = val;
            }
        }
    }
}

// ---------------------------------------------------------------------------
// Wave-per-edge: per-head dot(K[src], Q[dst])/sqrt(d) + basic_attn*W_dis[h].
// Stores logits[E,H] and does ordered-uint atomicMax segment max per (dst,h).
// ---------------------------------------------------------------------------
__global__ void edge_logits_kernel(const float* __restrict__ Kmat,
                                   const float* __restrict__ Qmat,
                                   const int* __restrict__ src,
                                   const int* __restrict__ dst,
                                   const float* __restrict__ basic_attn,
                                   const float* __restrict__ W_dis,
                                   float* __restrict__ logits,
                                   unsigned* __restrict__ mx_bits, int E_) {
    const int wid  = (blockIdx.x * blockDim.x + threadIdx.x) >> 5;
    const int lane = threadIdx.x & 31;
    if (wid >= E_) return;
    const int s = src[wid];
    const int t = dst[wid];

    const float4 kv = *(const float4*)&Kmat[(size_t)s * D_DIM + lane * 4];
    const float4 qv = *(const float4*)&Qmat[(size_t)t * D_DIM + lane * 4];
    float p = kv.x * qv.x + kv.y * qv.y + kv.z * qv.z + kv.w * qv.w;
    // head h = lane>>2 spans lanes 4h..4h+3 (16 floats)
    p += __shfl_xor(p, 1, 32);
    p += __shfl_xor(p, 2, 32);

    if ((lane & 3) == 0) {
        const int h = lane >> 2;
        const float logit = p * 0.25f + basic_attn[wid] * W_dis[h]; // 1/sqrt(16)
        logits[(size_t)wid * H_HEADS + h] = logit;
        atomicMax(&mx_bits[(size_t)t * H_HEADS + h], fflip(logit));
    }
}

// decode segment max in place; nodes with no in-edges (or -inf) -> 0 (ref fixup)
__global__ void mx_decode_kernel(unsigned* __restrict__ mx, int total) {
    const int i = blockIdx.x * blockDim.x + threadIdx.x;
    if (i >= total) return;
    const unsigned u = mx[i];
    float f = 0.0f;
    if (u != 0u) {
        f = funflip(u);
        if (!isfinite(f)) f = 0.0f;
    }
    ((float*)mx)[i] = f;
}

// thread per (edge,head): e = exp(logit - mx[dst,h]); accumulate denominator
__global__ void edge_exp_kernel(float* __restrict__ logits,
                                const float* __restrict__ mxf,
                                float* __restrict__ den,
                                const int* __restrict__ dst, int total) {
    const int i = blockIdx.x * blockDim.x + threadIdx.x;
    if (i >= total) return;
    const int e = i >> 3;
    const int h = i & 7;
    const int t = dst[e];
    const float ev = expf(logits[i] - mxf[(size_t)t * H_HEADS + h]);
    logits[i] = ev;  // overwrite logits with numerator
    atomicAdd(&den[(size_t)t * H_HEADS + h], ev);
}

// wave-per-edge: ft[dst] += V[e] * att[e,h]   (f32 global atomics)
__global__ void edge_scatter_kernel(const float* __restrict__ Vmat,
                                    const float* __restrict__ enum_buf,
                                    const float* __restrict__ den,
                                    const int* __restrict__ dst,
                                    float* __restrict__ ft, int E_) {
    const int wid  = (blockIdx.x * blockDim.x + threadIdx.x) >> 5;
    const int lane = threadIdx.x & 31;
    if (wid >= E_) return;
    const int t = dst[wid];
    const int h = lane >> 2;
    const float att = enum_buf[(size_t)wid * H_HEADS + h] /
                      (den[(size_t)t * H_HEADS + h] + 1e-16f);
    const float4 vv = *(const float4*)&Vmat[(size_t)wid * D_DIM + lane * 4];
    float* out = &ft[(size_t)t * D_DIM + lane * 4];
    atomicAdd(&out[0], vv.x * att);
    atomicAdd(&out[1], vv.y * att);
    atomicAdd(&out[2], vv.z * att);
    atomicAdd(&out[3], vv.w * att);
}

// wave-per-node: beta gate + LayerNorm1 -> he (residual path) and hln (FFN in)
__global__ void node_gate_ln_kernel(const float* __restrict__ ft,
                                    const float* __restrict__ x_emb,
                                    const float* __restrict__ W_beta,
                                    const float* __restrict__ g,
                                    const float* __restrict__ b,
                                    float* __restrict__ he_out,
                                    float* __restrict__ hln_out, int N_) {
    const int wid  = (blockIdx.x * blockDim.x + threadIdx.x) >> 5;
    const int lane = threadIdx.x & 31;
    if (wid >= N_) return;
    const size_t base = (size_t)wid * D_DIM + lane * 4;
    const int i0 = lane * 4;
    const float4 f = *(const float4*)&ft[base];
    const float4 x = *(const float4*)&x_emb[base];

    float p = 0.0f;
    p += f.x * W_beta[i0 + 0] + f.y * W_beta[i0 + 1] +
         f.z * W_beta[i0 + 2] + f.w * W_beta[i0 + 3];
    p += x.x * W_beta[128 + i0 + 0] + x.y * W_beta[128 + i0 + 1] +
         x.z * W_beta[128 + i0 + 2] + x.w * W_beta[128 + i0 + 3];
    p += (f.x - x.x) * W_beta[256 + i0 + 0] + (f.y - x.y) * W_beta[256 + i0 + 1] +
         (f.z - x.z) * W_beta[256 + i0 + 2] + (f.w - x.w) * W_beta[256 + i0 + 3];
    const float tot  = wave_sum32(p);
    const float beta = 1.0f / (1.0f + expf(-tot));

    float4 he;
    he.x = beta * x.x + (1.0f - beta) * f.x;
    he.y = beta * x.y + (1.0f - beta) * f.y;
    he.z = beta * x.z + (1.0f - beta) * f.z;
    he.w = beta * x.w + (1.0f - beta) * f.w;
    *(float4*)&he_out[base] = he;

    const float mu  = wave_sum32(he.x + he.y + he.z + he.w) * (1.0f / 128.0f);
    const float esq = wave_sum32(he.x * he.x + he.y * he.y +
                                 he.z * he.z + he.w * he.w) * (1.0f / 128.0f);
    const float inv = rsqrtf(esq - mu * mu + 1e-5f);

    float4 o;
    o.x = (he.x - mu) * inv * g[i0 + 0] + b[i0 + 0];
    o.y = (he.y - mu) * inv * g[i0 + 1] + b[i0 + 1];
    o.z = (he.z - mu) * inv * g[i0 + 2] + b[i0 + 2];
    o.w = (he.w - mu) * inv * g[i0 + 3] + b[i0 + 3];
    *(float4*)&hln_out[base] = o;
}

// wave-per-node: final residual + LayerNorm(ones, zeros) -> d_out
__global__ void final_ln_kernel(const float* __restrict__ ffn_out,
                                const float* __restrict__ he,
                                float* __restrict__ out, int N_) {
    const int wid  = (blockIdx.x * blockDim.x + threadIdx.x) >> 5;
    const int lane = threadIdx.x & 31;
    if (wid >= N_) return;
    const size_t base = (size_t)wid * D_DIM + lane * 4;
    const float4 a = *(const float4*)&ffn_out[base];
    const float4 r = *(const float4*)&he[base];
    float4 t;
    t.x = a.x + r.x; t.y = a.y + r.y; t.z = a.z + r.z; t.w = a.w + r.w;

    const float mu  = wave_sum32(t.x + t.y + t.z + t.w) * (1.0f / 128.0f);
    const float esq = wave_sum32(t.x * t.x + t.y * t.y +
                                 t.z * t.z + t.w * t.w) * (1.0f / 128.0f);
    const float inv = rsqrtf(esq - mu * mu + 1e-5f);

    float4 o;
    o.x = (t.x - mu) * inv;
    o.y = (t.y - mu) * inv;
    o.z = (t.z - mu) * inv;
    o.w = (t.w - mu) * inv;
    *(float4*)&out[base] = o;
}

// ---------------------------------------------------------------------------
// launch
// ---------------------------------------------------------------------------
extern "C" void kernel_launch(void* const* d_in, const int* in_sizes, int n_in,
                              void* d_out, int out_size, void* d_ws, size_t ws_size,
                              hipStream_t stream) {
    const float* node_emb   = (const float*)d_in[0];
    const float* bond_emb   = (const float*)d_in[1];
    const float* basic_attn = (const float*)d_in[2];
    const int*   src        = (const int*)d_in[3];
    const int*   dst        = (const int*)d_in[4];
    const float* Wk         = (const float*)d_in[5];
    const float* Wq         = (const float*)d_in[6];
    const float* Wv         = (const float*)d_in[7];
    const float* W_dis      = (const float*)d_in[8];
    const float* W_beta     = (const float*)d_in[9];
    const float* ln1_g      = (const float*)d_in[10];
    const float* ln1_b      = (const float*)d_in[11];
    const float* W_ff1      = (const float*)d_in[12];
    const float* W_ff2      = (const float*)d_in[13];

    const int N_ = in_sizes[0] / D_DIM;
    const int E_ = in_sizes[1] / D_DIM;

    // workspace layout (floats), with aliasing of dead buffers
    float*    Kmat   = (float*)d_ws;                         // [N,128]
    float*    Qmat   = Kmat + (size_t)N_ * D_DIM;            // [N,128]
    float*    Vmat   = Qmat + (size_t)N_ * D_DIM;            // [E,128]
    float*    logits = Vmat + (size_t)E_ * D_DIM;            // [E,8]
    unsigned* mx     = (unsigned*)(logits + (size_t)E_ * H_HEADS);  // [N,8]
    float*    mxf    = (float*)mx;                           // decoded in place
    float*    den    = mxf + (size_t)N_ * H_HEADS;           // [N,8]
    float*    ft     = den + (size_t)N_ * H_HEADS;           // [N,128]
    float*    he     = ft + (size_t)N_ * D_DIM;              // [N,128]
    float*    hln    = Kmat;   // K dead after edge_logits
    float*    ffo    = Qmat;   // Q dead after edge_logits
    float*    mid    = Vmat;   // V dead after edge_scatter ([N,256] < [E,128])

    // zero mx | den | ft in one contiguous memset (graph-capture safe)
    hipMemsetAsync(mx, 0,
                   ((size_t)N_ * H_HEADS * 2 + (size_t)N_ * D_DIM) * sizeof(float),
                   stream);

    const dim3 blk(256);

    // projections (WMMA f32)
    gemm_wmma_f32<<<dim3((N_ + 127) / 128, 1), blk, 0, stream>>>(
        node_emb, Wk, Kmat, N_, 128, 128, 0);
    gemm_wmma_f32<<<dim3((N_ + 127) / 128, 1), blk, 0, stream>>>(
        node_emb, Wq, Qmat, N_, 128, 128, 0);
    gemm_wmma_f32<<<dim3((E_ + 127) / 128, 1), blk, 0, stream>>>(
        bond_emb, Wv, Vmat, E_, 128, 128, 0);

    // edge softmax pipeline
    edge_logits_kernel<<<(E_ + 7) / 8, blk, 0, stream>>>(
        Kmat, Qmat, src, dst, basic_attn, W_dis, logits, mx, E_);
    mx_decode_kernel<<<(N_ * H_HEADS + 255) / 256, blk, 0, stream>>>(
        mx, N_ * H_HEADS);
    edge_exp_kernel<<<(E_ * H_HEADS + 255) / 256, blk, 0, stream>>>(
        logits, mxf, den, dst, E_ * H_HEADS);
    edge_scatter_kernel<<<(E_ + 7) / 8, blk, 0, stream>>>(
        Vmat, logits, den, dst, ft, E_);

    // gate + LN1
    node_gate_ln_kernel<<<(N_ + 7) / 8, blk, 0, stream>>>(
        ft, node_emb, W_beta, ln1_g, ln1_b, he, hln, N_);

    // FFN (WMMA f32, relu fused in first)
    gemm_wmma_f32<<<dim3((N_ + 127) / 128, 2), blk, 0, stream>>>(
        hln, W_ff1, mid, N_, 128, 256, 1);
    gemm_wmma_f32<<<dim3((N_ + 127) / 128, 1), blk, 0, stream>>>(
        mid, W_ff2, ffo, N_, 256, 128, 0);

    // residual + final LN
    final_ln_kernel<<<(N_ + 7) / 8, blk, 0, stream>>>(ffo, he, (float*)d_out, N_);
}